// Decoder_20555713479371
// MI455X (gfx1250) — compile-verified
//
#include <hip/hip_runtime.h>
#include <hip/hip_bf16.h>

// ---------------------------------------------------------------------------
// Fused masked-attention decoder step for MI455X (gfx1250, wave32).
// One workgroup per batch element b (512 blocks x 256 threads = 8 waves).
// Wave w == head h (H == 8).  Memory-bound: streams K, V, V_output once.
// Dot products on the two dominant streams run through
// V_WMMA_F32_16X16X4_F32 with the vector broadcast into B's 16 columns.
// ---------------------------------------------------------------------------

typedef __attribute__((ext_vector_type(2))) float v2f;
typedef __attribute__((ext_vector_type(8))) float v8f;

#define BDIM   512
#define GDIM   1000
#define HDIM   8
#define INDIM  256
#define EDIM   128
#define KDIM   16
#define GPAD   1008            // pad rows so 16-byte LDS alignment holds and
                               // tail tile (g0=992..1007) can spill harmlessly
#define NTILES 63              // ceil(1000/16)

__device__ __forceinline__ float waveMax(float v) {
  #pragma unroll
  for (int m = 16; m >= 1; m >>= 1) v = fmaxf(v, __shfl_xor(v, m, 32));
  return v;
}
__device__ __forceinline__ float waveSum(float v) {
  #pragma unroll
  for (int m = 16; m >= 1; m >>= 1) v += __shfl_xor(v, m, 32);
  return v;
}

__global__ __launch_bounds__(256)
void decoder_fused(const float* __restrict__ dec,      // (B,1,IN)
                   const float* __restrict__ K,        // (H,B,G,KD)
                   const float* __restrict__ V,        // (H,B,G,KD)
                   const float* __restrict__ VO,       // (B,G,E)
                   const int*   __restrict__ mask,     // (B,1,G)
                   const float* __restrict__ W_emb,    // (E,IN)
                   const float* __restrict__ b_emb,    // (E)
                   const float* __restrict__ W_query,  // (H,E,KD)
                   const float* __restrict__ W_out,    // (H,KD,E)
                   const float* __restrict__ W_q,      // (1,E,E)
                   float* __restrict__ out)            // (B,1,G)
{
  __shared__ float s_dec[INDIM];
  __shared__ float s_emb[EDIM];
  __shared__ float s_q[HDIM * KDIM];
  __shared__ float s_attn[HDIM][GPAD];
  __shared__ float s_heads[HDIM * KDIM];
  __shared__ float s_ctx[EDIM];
  __shared__ float s_q2[EDIM];

  const int b    = blockIdx.x;
  const int t    = threadIdx.x;
  const int wave = t >> 5;        // == head h
  const int lane = t & 31;

  // ---------------- Phase 0: embedding + per-head queries -----------------
  if (t < INDIM) s_dec[t] = dec[(size_t)b * INDIM + t];
  __syncthreads();

  if (t < EDIM) {
    float acc = b_emb[t];
    const float* we = W_emb + (size_t)t * INDIM;
    #pragma unroll 4
    for (int i = 0; i < INDIM; ++i) acc += s_dec[i] * we[i];
    s_emb[t] = acc;
  }
  __syncthreads();

  if (t < HDIM * KDIM) {
    const int h = t >> 4, kd = t & 15;
    float acc = 0.f;
    const float* wq = W_query + ((size_t)h * EDIM) * KDIM + kd;
    #pragma unroll 4
    for (int e = 0; e < EDIM; ++e) acc += s_emb[e] * wq[(size_t)e * KDIM];
    s_q[t] = acc;
  }
  __syncthreads();

  // ---------------- Phase 1: compat = Q . K  (WMMA f32 16x16x4) -----------
  // A = 16 rows of K (g-tile) x 4 kd;  B = Q_h broadcast into 16 columns.
  // D row M (replicated over N) = compat[g0+M].
  {
    const int h = wave;
    const float* Kh    = K + ((size_t)(h * BDIM + b)) * GDIM * KDIM;
    const int    mrow  = lane & 15;
    const int    khalf = (lane >> 4) << 1;   // lanes 0-15: K{0,1}; 16-31: K{2,3}

    for (int tI = 0; tI < NTILES; ++tI) {
      const int g0  = tI * 16;
      int       row = g0 + mrow;  if (row > GDIM - 1) row = GDIM - 1;  // tail clamp
      const float* krow = Kh + (size_t)row * KDIM + khalf;
      __builtin_prefetch(krow + 16 * KDIM, 0, 1);  // next g-tile

      v8f acc = {};
      #pragma unroll
      for (int c = 0; c < 4; ++c) {            // kd chunks of 4
        v2f a;  a.x  = krow[4 * c];            a.y  = krow[4 * c + 1];
        v2f bb; bb.x = s_q[h * KDIM + 4 * c + khalf];
                bb.y = s_q[h * KDIM + 4 * c + khalf + 1];
        acc = __builtin_amdgcn_wmma_f32_16x16x4_f32(
                  false, a, false, bb, (short)0, acc, false, false);
      }
      if (lane == 0) {
        #pragma unroll
        for (int r = 0; r < 8; ++r) s_attn[h][g0 + r] = acc[r];
      } else if (lane == 16) {
        #pragma unroll
        for (int r = 0; r < 8; ++r) s_attn[h][g0 + 8 + r] = acc[r];  // tail spills into pad
      }
    }
  }
  __syncthreads();

  // ---------------- Phase 2: masked softmax over g (per head wave) --------
  {
    const int h = wave;
    const int* mrow = mask + (size_t)b * GDIM;
    const float NEG_INF = -__builtin_inff();

    float mx = NEG_INF;
    for (int g = lane; g < GDIM; g += 32) {
      float v = s_attn[h][g] * 0.25f;               // 1/sqrt(KD)
      v = (mrow[g] != 0) ? v : NEG_INF;             // mask==0 -> forbidden
      s_attn[h][g] = v;
      mx = fmaxf(mx, v);
    }
    mx = waveMax(mx);

    float sum = 0.f;
    for (int g = lane; g < GDIM; g += 32) {
      float v = s_attn[h][g];
      float e = (v == NEG_INF) ? 0.f : __expf(v - mx);
      s_attn[h][g] = e;
      sum += e;
    }
    sum = waveSum(sum);
    const float inv = (sum > 0.f) ? 1.f / sum : 0.f;  // all-masked -> attn 0 (matches ref)
    for (int g = lane; g < GDIM; g += 32) s_attn[h][g] *= inv;
  }
  __syncthreads();

  // ---------------- Phase 3: heads_h = sum_g attn[g] * V[h,b,g,:] ---------
  // 8 rows per iteration, float4 per lane -> fully coalesced 512B/wave/iter.
  {
    const int h = wave;
    const float* Vh = V + ((size_t)(h * BDIM + b)) * GDIM * KDIM;
    const int rofs = lane >> 2;   // 0..7 : row within group of 8
    const int q    = lane & 3;    // kd quad
    float4 acc = make_float4(0.f, 0.f, 0.f, 0.f);

    for (int i = 0; i < GDIM / 8; ++i) {          // 125 iterations exactly
      const int g = i * 8 + rofs;
      const float a = s_attn[h][g];
      const float4 vv = *reinterpret_cast<const float4*>(Vh + (size_t)g * KDIM + q * 4);
      acc.x += a * vv.x; acc.y += a * vv.y; acc.z += a * vv.z; acc.w += a * vv.w;
    }
    #pragma unroll
    for (int m = 4; m <= 16; m <<= 1) {
      acc.x += __shfl_xor(acc.x, m, 32);
      acc.y += __shfl_xor(acc.y, m, 32);
      acc.z += __shfl_xor(acc.z, m, 32);
      acc.w += __shfl_xor(acc.w, m, 32);
    }
    if (lane < 4) {
      s_heads[h * KDIM + q * 4 + 0] = acc.x;
      s_heads[h * KDIM + q * 4 + 1] = acc.y;
      s_heads[h * KDIM + q * 4 + 2] = acc.z;
      s_heads[h * KDIM + q * 4 + 3] = acc.w;
    }
  }
  __syncthreads();

  // ---------------- Phase 4: context = heads @ W_out ; Q2 = context @ W_q -
  if (t < EDIM) {
    float acc = 0.f;
    #pragma unroll 4
    for (int hk = 0; hk < HDIM * KDIM; ++hk)
      acc += s_heads[hk] * W_out[(size_t)hk * EDIM + t];
    s_ctx[t] = acc;
  }
  __syncthreads();
  if (t < EDIM) {
    float acc = 0.f;
    #pragma unroll 4
    for (int e = 0; e < EDIM; ++e)
      acc += s_ctx[e] * W_q[(size_t)e * EDIM + t];
    s_q2[t] = acc;
  }
  __syncthreads();

  // ---------------- Phase 5: z = Q2 . V_output / sqrt(E); out = 10 tanh(z)
  // Same WMMA broadcast trick: A = 16 rows of V_output x 4 e; 32 chunks.
  {
    const float invSqrtE = 0.08838834764831845f;   // 1/sqrt(128)
    const float* vob  = VO + (size_t)b * GDIM * EDIM;
    const int    mrow = lane & 15;
    const int    khalf = (lane >> 4) << 1;

    for (int tI = wave; tI < NTILES; tI += HDIM) {
      const int g0  = tI * 16;
      int       row = g0 + mrow;  if (row > GDIM - 1) row = GDIM - 1;
      const float* vr = vob + (size_t)row * EDIM + khalf;
      __builtin_prefetch(vr + (size_t)HDIM * 16 * EDIM, 0, 1);  // this wave's next tile

      v8f acc = {};
      #pragma unroll
      for (int c = 0; c < 32; ++c) {            // e chunks of 4
        v2f a;  a.x  = vr[4 * c];             a.y  = vr[4 * c + 1];
        v2f bb; bb.x = s_q2[4 * c + khalf];   bb.y = s_q2[4 * c + khalf + 1];
        acc = __builtin_amdgcn_wmma_f32_16x16x4_f32(
                  false, a, false, bb, (short)0, acc, false, false);
      }
      if (lane == 0 || lane == 16) {
        const int gb = g0 + ((lane >> 4) << 3);
        #pragma unroll
        for (int r = 0; r < 8; ++r) {
          const int g = gb + r;
          if (g < GDIM)
            out[(size_t)b * GDIM + g] = 10.f * tanhf(acc[r] * invSqrtE);
        }
      }
    }
  }
}

extern "C" void kernel_launch(void* const* d_in, const int* in_sizes, int n_in,
                              void* d_out, int out_size, void* d_ws, size_t ws_size,
                              hipStream_t stream) {
  const float* dec     = (const float*)d_in[0];
  const float* K       = (const float*)d_in[1];
  const float* V       = (const float*)d_in[2];
  const float* VO      = (const float*)d_in[3];
  const int*   mask    = (const int*)  d_in[4];
  const float* W_emb   = (const float*)d_in[5];
  const float* b_emb   = (const float*)d_in[6];
  const float* W_query = (const float*)d_in[7];
  const float* W_out   = (const float*)d_in[8];
  const float* W_q     = (const float*)d_in[9];
  float* out = (float*)d_out;

  decoder_fused<<<BDIM, 256, 0, stream>>>(dec, K, V, VO, mask,
                                          W_emb, b_emb, W_query, W_out, W_q, out);
}